// ProximityAttention_54580444397661
// MI455X (gfx1250) — compile-verified
//
#include <hip/hip_runtime.h>

typedef float v2f __attribute__((ext_vector_type(2)));
typedef float v8f __attribute__((ext_vector_type(8)));

#define NNODES 2048
#define NTILES 128          // NNODES/16
#define NHEADS 8
#define KQ_SCALE 0.1f
#define LDSTRIDE 20         // 80B rows: 16B-aligned for async b128, conflict-free

#if defined(__AMDGCN__) && __has_builtin(__builtin_amdgcn_global_load_async_to_lds_b128) && \
    __has_builtin(__builtin_amdgcn_global_load_async_to_lds_b32) &&                         \
    __has_builtin(__builtin_amdgcn_s_wait_asynccnt)
#define USE_ASYNC 1
typedef int i32x4 __attribute__((__vector_size__(16)));
typedef __attribute__((address_space(1))) i32x4 g_i32x4;
typedef __attribute__((address_space(3))) i32x4 l_i32x4;
typedef __attribute__((address_space(1))) int g_i32;
typedef __attribute__((address_space(3))) int l_i32;
#else
#define USE_ASYNC 0
#endif

// D(16x16,f32) = A(16x4,f32) * B(4x16,f32) + C
// A: lane l, vgpr j -> A[M=l%16, K=j+2*(l/16)]
// B: lane l, vgpr j -> B[K=j+2*(l/16), N=l%16]
// C/D: vgpr r, lane l -> [M=r+8*(l/16), N=l%16]
__device__ inline v8f wmma4(v2f a, v2f b, v8f c) {
  return __builtin_amdgcn_wmma_f32_16x16x4_f32(false, a, false, b, (short)0, c,
                                               false, false);
}

// ---------------- Kernel 1: per-head projections (WMMA GEMMs) ----------------
__device__ inline v8f proj_tile(const float* __restrict__ W, int ldW, int nrows,
                                const float* __restrict__ X, int IN, int lane) {
  int col = lane & 15;
  int row = (col < nrows) ? col : (nrows - 1);  // clamp padded rows in-bounds
  int kh = (lane >> 4) << 1;
  v8f acc = {0.f, 0.f, 0.f, 0.f, 0.f, 0.f, 0.f, 0.f};
  for (int c0 = 0; c0 < IN; c0 += 4) {
    v2f a, b;
    a.x = W[row * ldW + c0 + kh];
    a.y = W[row * ldW + c0 + kh + 1];
    b.x = X[(c0 + kh) * 17 + col];
    b.y = X[(c0 + kh + 1) * 17 + col];
    acc = wmma4(a, b, acc);
  }
  return acc;
}

__device__ inline void store_feat(float* __restrict__ dst, int featBase, v8f d,
                                  int nrows, float scale, int lane) {
  int colN = lane & 15;
  int rb = (lane >> 4) * 8;
#pragma unroll
  for (int r = 0; r < 8; ++r) {
    int row = rb + r;
    if (row < nrows) dst[(featBase + row) * 16 + colN] = d[r] * scale;
  }
}

__global__ __launch_bounds__(32) void prox_proj_kernel(
    const float* __restrict__ ax, const float* __restrict__ vx,
    const float* __restrict__ W_ak, const float* __restrict__ W_vk,
    const float* __restrict__ W_aq, const float* __restrict__ W_vq,
    const float* __restrict__ W_aval, const float* __restrict__ W_vval,
    const float* __restrict__ W_a2vval, const float* __restrict__ W_v2aval,
    float* __restrict__ KFs, float* __restrict__ QFs, float* __restrict__ VFs) {
  __shared__ float xa[128 * 17];      // [c][node], stride 17
  __shared__ float xv[3 * 64 * 17];   // [v][c][node]
  int tile = blockIdx.x, h = blockIdx.y, lane = threadIdx.x;
  int n0 = tile * 16;

  for (int i = 0; i < 64; ++i) {      // ax tile: 16 nodes x 128 ch
    int t = i * 32 + lane;
    xa[(t & 127) * 17 + (t >> 7)] = ax[n0 * 128 + t];
  }
  for (int i = 0; i < 96; ++i) {      // vx tile: 16 nodes x 64 ch x 3
    int t = i * 32 + lane;
    int node = t / 192, rem = t - node * 192;
    xv[((rem % 3) * 64 + rem / 3) * 17 + node] = vx[n0 * 192 + t];
  }
  __syncthreads();

  float* kf = KFs + (h * NTILES + tile) * 80 * 16;
  float* qf = QFs + (h * NTILES + tile) * 80 * 16;
  float* vf = VFs + (h * NTILES + tile) * 96 * 16;

  // KF = [0.1*ak(32), 0.1*vk_v(3*16)]  (KQ_SCALE folded into key side)
  store_feat(kf, 0,  proj_tile(W_ak + (h * 32 + 0)  * 128, 128, 16, xa, 128, lane), 16, KQ_SCALE, lane);
  store_feat(kf, 16, proj_tile(W_ak + (h * 32 + 16) * 128, 128, 16, xa, 128, lane), 16, KQ_SCALE, lane);
  for (int v = 0; v < 3; ++v)
    store_feat(kf, 32 + 16 * v, proj_tile(W_vk + h * 16 * 64, 64, 16, xv + v * 64 * 17, 64, lane), 16, KQ_SCALE, lane);
  // QF = [aq(32), vq_v(3*16)]
  store_feat(qf, 0,  proj_tile(W_aq + (h * 32 + 0)  * 128, 128, 16, xa, 128, lane), 16, 1.f, lane);
  store_feat(qf, 16, proj_tile(W_aq + (h * 32 + 16) * 128, 128, 16, xa, 128, lane), 16, 1.f, lane);
  for (int v = 0; v < 3; ++v)
    store_feat(qf, 32 + 16 * v, proj_tile(W_vq + h * 16 * 64, 64, 16, xv + v * 64 * 17, 64, lane), 16, 1.f, lane);
  // VF = [avalT(16), v2aT_v(3*16), vvalT_v(3*8), a2vT(8)]
  store_feat(vf, 0, proj_tile(W_aval + h * 16 * 128, 128, 16, xa, 128, lane), 16, 1.f, lane);
  for (int v = 0; v < 3; ++v)
    store_feat(vf, 16 + 16 * v, proj_tile(W_v2aval + h * 16 * 64, 64, 16, xv + v * 64 * 17, 64, lane), 16, 1.f, lane);
  for (int v = 0; v < 3; ++v)
    store_feat(vf, 64 + 8 * v, proj_tile(W_vval + h * 8 * 64, 64, 8, xv + v * 64 * 17, 64, lane), 8, 1.f, lane);
  store_feat(vf, 88, proj_tile(W_a2vval + h * 8 * 128, 128, 8, xa, 128, lane), 8, 1.f, lane);
}

// ---------------- Kernel 2: flash attention with proximity bias --------------
__global__ __launch_bounds__(128) void prox_attn_kernel(
    const float* __restrict__ KFs, const float* __restrict__ QFs,
    const float* __restrict__ VFs, const float* __restrict__ pos_k,
    const float* __restrict__ pos_q, const float* __restrict__ r0,
    float* __restrict__ out_a, float* __restrict__ out_v) {
  __shared__ float lds_kf[2][80 * LDSTRIDE];
  __shared__ float lds_vf[2][96 * LDSTRIDE];
  __shared__ float lds_pos[2][64];
  __shared__ float lds_p[4 * 4 * 272];  // per-wave: P, P*dx, P*dy, P*dz (16x16 @ 17)

  int h = blockIdx.y;
  int tid = threadIdx.x;
  int wave = tid >> 5, lane = tid & 31;
  int col = lane & 15;       // query within wave's tile / node column
  int hi = lane >> 4;        // 0/1 half
  int col7 = col & 7;
  int qtile = blockIdx.x * 4 + wave;
  int m = qtile * 16 + col;  // global output node index
  float* myP = &lds_p[wave * 4 * 272];

  const float* kf0 = KFs + h * NTILES * 80 * 16;
  const float* vf0 = VFs + h * NTILES * 96 * 16;

  // QF for this wave's 16 queries, in B-matrix layout (20 K-chunks of 4)
  const float* qfp = QFs + (h * NTILES + qtile) * 80 * 16;
  v2f qreg[20];
#pragma unroll
  for (int c = 0; c < 20; ++c) {
    qreg[c].x = qfp[(4 * c + 2 * hi) * 16 + col];
    qreg[c].y = qfp[(4 * c + 2 * hi + 1) * 16 + col];
  }
  // output axis m pairs with pos_k (reference broadcasting quirk)
  float pqx = pos_k[m * 3 + 0], pqy = pos_k[m * 3 + 1], pqz = pos_k[m * 3 + 2];
  float r0h = r0[h], r0sq = r0h * r0h, logr0sq = __logf(r0sq);

  float run_max = -3.0e38f, run_sum = 0.f;
  v8f acc_a = {0.f, 0.f, 0.f, 0.f, 0.f, 0.f, 0.f, 0.f};
  v8f acc_v0 = acc_a, acc_v1 = acc_a, acc_v2 = acc_a;

  // ---- prologue: stage tile 0 into buffer 0 ----
#if USE_ASYNC
  for (int q = tid; q < 320; q += 128) {  // KF: 320 x 16B chunks
    int f = q >> 2, p = q & 3;
    __builtin_amdgcn_global_load_async_to_lds_b128(
        (g_i32x4*)(kf0 + q * 4), (l_i32x4*)&lds_kf[0][f * LDSTRIDE + p * 4], 0, 0);
  }
  for (int q = tid; q < 384; q += 128) {  // VF: 384 x 16B chunks
    int f = q >> 2, p = q & 3;
    __builtin_amdgcn_global_load_async_to_lds_b128(
        (g_i32x4*)(vf0 + q * 4), (l_i32x4*)&lds_vf[0][f * LDSTRIDE + p * 4], 0, 0);
  }
  if (tid < 48)
    __builtin_amdgcn_global_load_async_to_lds_b32(
        (g_i32*)(pos_q + tid), (l_i32*)&lds_pos[0][(tid / 3) * 4 + tid % 3], 0, 0);
  __builtin_amdgcn_s_wait_asynccnt(0);
#else
  for (int i = tid; i < 80 * 16; i += 128)
    lds_kf[0][(i >> 4) * LDSTRIDE + (i & 15)] = kf0[i];
  for (int i = tid; i < 96 * 16; i += 128)
    lds_vf[0][(i >> 4) * LDSTRIDE + (i & 15)] = vf0[i];
  if (tid < 48) lds_pos[0][(tid / 3) * 4 + tid % 3] = pos_q[tid];
#endif
  __syncthreads();

  for (int kt = 0; kt < NTILES; ++kt) {
    int cur = kt & 1, nxt = cur ^ 1;
#if !USE_ASYNC
    float rk[10], rv[12], rp = 0.f;
#endif
    // ---- issue staging of tile kt+1 into the other buffer (overlaps WMMAs) ----
    if (kt + 1 < NTILES) {
      const float* kfp = kf0 + (kt + 1) * 80 * 16;
      const float* vfp = vf0 + (kt + 1) * 96 * 16;
      const float* ppp = pos_q + (kt + 1) * 48;
#if USE_ASYNC
      for (int q = tid; q < 320; q += 128) {
        int f = q >> 2, p = q & 3;
        __builtin_amdgcn_global_load_async_to_lds_b128(
            (g_i32x4*)(kfp + q * 4), (l_i32x4*)&lds_kf[nxt][f * LDSTRIDE + p * 4], 0, 0);
      }
      for (int q = tid; q < 384; q += 128) {
        int f = q >> 2, p = q & 3;
        __builtin_amdgcn_global_load_async_to_lds_b128(
            (g_i32x4*)(vfp + q * 4), (l_i32x4*)&lds_vf[nxt][f * LDSTRIDE + p * 4], 0, 0);
      }
      if (tid < 48)
        __builtin_amdgcn_global_load_async_to_lds_b32(
            (g_i32*)(ppp + tid), (l_i32*)&lds_pos[nxt][(tid / 3) * 4 + tid % 3], 0, 0);
#else
#pragma unroll
      for (int i = 0; i < 10; ++i) rk[i] = kfp[tid + 128 * i];
#pragma unroll
      for (int i = 0; i < 12; ++i) rv[i] = vfp[tid + 128 * i];
      if (tid < 48) rp = ppp[tid];
#endif
    }
    if (kt + 2 < NTILES) {  // warm L2/WGP$ for the tile after next
      __builtin_prefetch(kf0 + (kt + 2) * 1280 + tid * 10, 0, 1);
      __builtin_prefetch(vf0 + (kt + 2) * 1536 + tid * 12, 0, 1);
    }

    const float* kfb = lds_kf[cur];
    const float* vfb = lds_vf[cur];
    const float* psb = lds_pos[cur];

    // scores S[key, query] = sum_f KF[f,key]*QF[f,query] (K=80, split chains)
    v8f sA = {0.f, 0.f, 0.f, 0.f, 0.f, 0.f, 0.f, 0.f};
    v8f sB = sA;
#pragma unroll
    for (int c = 0; c < 10; ++c) {
      v2f a;
      a.x = kfb[(8 * c + 2 * hi) * LDSTRIDE + col];
      a.y = kfb[(8 * c + 2 * hi + 1) * LDSTRIDE + col];
      sA = wmma4(a, qreg[2 * c], sA);
      a.x = kfb[(8 * c + 4 + 2 * hi) * LDSTRIDE + col];
      a.y = kfb[(8 * c + 4 + 2 * hi + 1) * LDSTRIDE + col];
      sB = wmma4(a, qreg[2 * c + 1], sB);
    }

    // elementwise proximity bias + online softmax
    float logit[8], dx[8], dy[8], dz[8];
    float lmax = -3.0e38f;
#pragma unroll
    for (int r = 0; r < 8; ++r) {
      int k = r + 8 * hi;
      float sx = pqx - psb[k * 4 + 0];
      float sy = pqy - psb[k * 4 + 1];
      float sz = pqz - psb[k * 4 + 2];
      float den = r0sq + sx * sx + sy * sy + sz * sz;
      logit[r] = (sA[r] + sB[r]) + logr0sq - __logf(den);
      float rs = rsqrtf(den);  // sqrt(prox/r0^2) = 1/sqrt(r0^2+d^2)
      dx[r] = sx * rs; dy[r] = sy * rs; dz[r] = sz * rs;
      lmax = fmaxf(lmax, logit[r]);
    }
    lmax = fmaxf(lmax, __shfl_xor(lmax, 16, 32));
    float nmax = fmaxf(run_max, lmax);
    float rescale = __expf(run_max - nmax);
    run_max = nmax;
    float lsum = 0.f;
#pragma unroll
    for (int r = 0; r < 8; ++r) {
      int k = r + 8 * hi;
      float pe = __expf(logit[r] - nmax);
      lsum += pe;
      myP[0 * 272 + k * 17 + col] = pe;
      myP[1 * 272 + k * 17 + col] = pe * dx[r];
      myP[2 * 272 + k * 17 + col] = pe * dy[r];
      myP[3 * 272 + k * 17 + col] = pe * dz[r];
    }
    lsum += __shfl_xor(lsum, 16, 32);
    run_sum = run_sum * rescale + lsum;
#pragma unroll
    for (int i = 0; i < 8; ++i) {
      acc_a[i] *= rescale; acc_v0[i] *= rescale;
      acc_v1[i] *= rescale; acc_v2[i] *= rescale;
    }

    // re-materialize P / P_v tiles as B-matrix chunks (same-wave LDS: in-order)
    v2f pb[4], p0[4], p1[4], p2[4];
#pragma unroll
    for (int c = 0; c < 4; ++c) {
      int k = 4 * c + 2 * hi;
      pb[c].x = myP[0 * 272 + k * 17 + col]; pb[c].y = myP[0 * 272 + (k + 1) * 17 + col];
      p0[c].x = myP[1 * 272 + k * 17 + col]; p0[c].y = myP[1 * 272 + (k + 1) * 17 + col];
      p1[c].x = myP[2 * 272 + k * 17 + col]; p1[c].y = myP[2 * 272 + (k + 1) * 17 + col];
      p2[c].x = myP[3 * 272 + k * 17 + col]; p2[c].y = myP[3 * 272 + (k + 1) * 17 + col];
    }
    // value matmuls: 40 WMMAs per key tile
#pragma unroll
    for (int c = 0; c < 4; ++c) {
      int k = 4 * c + 2 * hi;
      v2f a;
      a.x = vfb[(0 + col) * LDSTRIDE + k];  a.y = vfb[(0 + col) * LDSTRIDE + k + 1];
      acc_a = wmma4(a, pb[c], acc_a);                       // aval
      a.x = vfb[(16 + col) * LDSTRIDE + k]; a.y = vfb[(16 + col) * LDSTRIDE + k + 1];
      acc_a = wmma4(a, p0[c], acc_a);                       // v2a . dir
      a.x = vfb[(32 + col) * LDSTRIDE + k]; a.y = vfb[(32 + col) * LDSTRIDE + k + 1];
      acc_a = wmma4(a, p1[c], acc_a);
      a.x = vfb[(48 + col) * LDSTRIDE + k]; a.y = vfb[(48 + col) * LDSTRIDE + k + 1];
      acc_a = wmma4(a, p2[c], acc_a);
      a.x = vfb[(64 + col7) * LDSTRIDE + k]; a.y = vfb[(64 + col7) * LDSTRIDE + k + 1];
      acc_v0 = wmma4(a, pb[c], acc_v0);                     // vval_v
      a.x = vfb[(72 + col7) * LDSTRIDE + k]; a.y = vfb[(72 + col7) * LDSTRIDE + k + 1];
      acc_v1 = wmma4(a, pb[c], acc_v1);
      a.x = vfb[(80 + col7) * LDSTRIDE + k]; a.y = vfb[(80 + col7) * LDSTRIDE + k + 1];
      acc_v2 = wmma4(a, pb[c], acc_v2);
      a.x = vfb[(88 + col7) * LDSTRIDE + k]; a.y = vfb[(88 + col7) * LDSTRIDE + k + 1];
      acc_v0 = wmma4(a, p0[c], acc_v0);                     // a2v . dir_v
      acc_v1 = wmma4(a, p1[c], acc_v1);
      acc_v2 = wmma4(a, p2[c], acc_v2);
    }

    // ---- finish staging of tile kt+1, publish to all waves ----
#if USE_ASYNC
    __builtin_amdgcn_s_wait_asynccnt(0);
#else
    if (kt + 1 < NTILES) {
#pragma unroll
      for (int i = 0; i < 10; ++i) {
        int t = tid + 128 * i;
        lds_kf[nxt][(t >> 4) * LDSTRIDE + (t & 15)] = rk[i];
      }
#pragma unroll
      for (int i = 0; i < 12; ++i) {
        int t = tid + 128 * i;
        lds_vf[nxt][(t >> 4) * LDSTRIDE + (t & 15)] = rv[i];
      }
      if (tid < 48) lds_pos[nxt][(tid / 3) * 4 + tid % 3] = rp;
    }
#endif
    __syncthreads();
  }

  float inv = 1.0f / run_sum;
#pragma unroll
  for (int r = 0; r < 8; ++r)
    out_a[m * 128 + h * 16 + r + 8 * hi] = acc_a[r] * inv;  // aans[m, h*16+i]
  if (hi == 0) {  // acc_v rows 0..7 valid (i < 8)
#pragma unroll
    for (int r = 0; r < 8; ++r) {
      int cch = h * 8 + r;
      out_v[m * 192 + cch * 3 + 0] = acc_v0[r] * inv;
      out_v[m * 192 + cch * 3 + 1] = acc_v1[r] * inv;
      out_v[m * 192 + cch * 3 + 2] = acc_v2[r] * inv;
    }
  }
}

extern "C" void kernel_launch(void* const* d_in, const int* in_sizes, int n_in,
                              void* d_out, int out_size, void* d_ws, size_t ws_size,
                              hipStream_t stream) {
  const float* ax       = (const float*)d_in[0];
  const float* vx       = (const float*)d_in[1];
  const float* pos_k    = (const float*)d_in[2];
  const float* pos_q    = (const float*)d_in[3];
  const float* r0       = (const float*)d_in[4];
  const float* W_ak     = (const float*)d_in[5];
  const float* W_vk     = (const float*)d_in[6];
  const float* W_aq     = (const float*)d_in[7];
  const float* W_vq     = (const float*)d_in[8];
  const float* W_aval   = (const float*)d_in[9];
  const float* W_vval   = (const float*)d_in[10];
  const float* W_a2vval = (const float*)d_in[11];
  const float* W_v2aval = (const float*)d_in[12];
  float* out = (float*)d_out;
  float* ws = (float*)d_ws;

  // scratch: KF (8*128*80*16) + QF (same) + VF (8*128*96*16) = 16.8 MB (L2-resident)
  float* KFs = ws;
  float* QFs = KFs + NHEADS * NTILES * 80 * 16;
  float* VFs = QFs + NHEADS * NTILES * 80 * 16;

  prox_proj_kernel<<<dim3(NTILES, NHEADS), 32, 0, stream>>>(
      ax, vx, W_ak, W_vk, W_aq, W_vq, W_aval, W_vval, W_a2vval, W_v2aval,
      KFs, QFs, VFs);
  prox_attn_kernel<<<dim3(NTILES / 4, NHEADS), 128, 0, stream>>>(
      KFs, QFs, VFs, pos_k, pos_q, r0, out, out + NNODES * 128);
}